// PointNet_V3_75033078661467
// MI455X (gfx1250) — compile-verified
//
#include <hip/hip_runtime.h>

typedef float v2f __attribute__((ext_vector_type(2)));
typedef float v8f __attribute__((ext_vector_type(8)));

#define PN_EPS 1e-3f

// ---------------- LDS layout (floats) ----------------
// Padded strides chosen so the two lane-halves (which read rows k and k+2)
// hit disjoint 16-bank windows: (2*stride) % 64 >= 16.
#define W1S   72      // W1 folded:  [16][72]  (cols 0..63 used)
#define W2AS  136     // W2a folded: [16][136] (cols 0..127 used)
#define W2BS  72      // W2b folded: [128][72] (cols 0..63 used)
#define HS    20      // h staging tile per wave: [16][20] (cols 0..15 used)

#define OFF_W1F   0
#define OFF_B1F   (OFF_W1F  + 16*W1S)       // 1152
#define OFF_W2AF  (OFF_B1F  + 64)           // 1216
#define OFF_B2AF  (OFF_W2AF + 16*W2AS)      // 3392
#define OFF_W2BF  (OFF_B2AF + 128)          // 3520
#define OFF_B2BF  (OFF_W2BF + 128*W2BS)     // 12736
#define OFF_HST   (OFF_B2BF + 64)           // 12800
#define LDS_TOTAL (OFF_HST  + 8*16*HS)      // 15360 floats = 61440 B

__device__ __forceinline__ v8f wmma4(v2f a, v2f b, v8f c) {
  // D = A(16x4,f32) * B(4x16,f32) + C(16x16,f32)
  return __builtin_amdgcn_wmma_f32_16x16x4_f32(false, a, false, b, (short)0, c,
                                               false, false);
}

// ---------------- zero accumulators ----------------
__global__ void pn_zero(float* __restrict__ out, int n, float* __restrict__ gmax) {
  int i = blockIdx.x * blockDim.x + threadIdx.x;
  if (i < n) __builtin_nontemporal_store(0.0f, out + i);
  if (blockIdx.x == 0 && threadIdx.x < 64) gmax[threadIdx.x] = 0.0f;
}

// ---------------- per-tile fused work ----------------
// FULL=true: all 16 rows of the tile are valid (wave-uniform hot path, no
// per-element guards -> straight-line atomics, hoisted segment ids).
template <bool FULL>
__device__ __forceinline__ void pn_tile(
    const float* __restrict__ in, const int* __restrict__ seg,
    float* __restrict__ out, float* __restrict__ gmax,
    float* lds, float* hst, int r0, int N,
    int lane, int half, int lm)
{
  // --- A fragments of the input tile (16x16 f32, four K=4 chunks)
  v2f a[4];
  {
    int arow = r0 + lm;
    const float* ap = in + (size_t)arow * 16 + 2 * half;
    if (FULL || arow < N) {
#pragma unroll
      for (int q = 0; q < 4; ++q)
        a[q] = __builtin_nontemporal_load((const v2f*)(ap + 4 * q));
    } else {
      const v2f zero2 = {0.0f, 0.0f};
#pragma unroll
      for (int q = 0; q < 4; ++q) a[q] = zero2;
    }
  }

  // --- segment ids: one coalesced load + lane permutes
  // lanes 0..15 load seg[r0+lane]; lane-half needs rows half*8 + j.
  int  sid[8]; bool rv[8];
  {
    int r = r0 + lane;
    int v = 0;
    if (lane < 16 && (FULL || r < N)) v = seg[r];
#pragma unroll
    for (int j = 0; j < 8; ++j) {
      sid[j] = __shfl(v, half * 8 + j, 32);
      rv[j]  = FULL || (r0 + half * 8 + j) < N;
    }
  }

  int* outi = (int*)out;

  // --- Branch 1: x = relu(bn(in @ W1)); scatter sum + max per segment
#pragma unroll
  for (int n = 0; n < 4; ++n) {
    const int col = n * 16 + lm;
    float bb = lds[OFF_B1F + col];
    v8f acc;
#pragma unroll
    for (int j = 0; j < 8; ++j) acc[j] = bb;
#pragma unroll
    for (int q = 0; q < 4; ++q) {
      int kr = 4 * q + 2 * half;
      v2f b = { lds[OFF_W1F + kr * W1S + col],
                lds[OFF_W1F + (kr + 1) * W1S + col] };
      acc = wmma4(a[q], b, acc);
    }
#pragma unroll
    for (int j = 0; j < 8; ++j) {
      float v = acc[j] > 0.0f ? acc[j] : 0.0f;
      if (FULL || rv[j]) {
        int base = sid[j] * 192 + col * 3;             // channel-shuffled layout
        atomicAdd(out + base, v);                      // group 0: segment sum
        atomicMax(outi + base + 1, __float_as_int(v)); // group 1: segment max (v>=0)
      }
    }
  }

  // --- Branch 2: h = relu(bn(in @ W2a)) ; x2 = relu(bn(h @ W2b)) ; global max
  v8f acc2[4];
#pragma unroll
  for (int n = 0; n < 4; ++n) {
    float bb = lds[OFF_B2BF + n * 16 + lm];
#pragma unroll
    for (int j = 0; j < 8; ++j) acc2[n][j] = bb;
  }

#pragma unroll
  for (int n2 = 0; n2 < 8; ++n2) {
    const int col = n2 * 16 + lm;
    float bb = lds[OFF_B2AF + col];
    v8f h;
#pragma unroll
    for (int j = 0; j < 8; ++j) h[j] = bb;
#pragma unroll
    for (int q = 0; q < 4; ++q) {
      int kr = 4 * q + 2 * half;
      v2f b = { lds[OFF_W2AF + kr * W2AS + col],
                lds[OFF_W2AF + (kr + 1) * W2AS + col] };
      h = wmma4(a[q], b, h);
    }
    // relu + stage this 16x16 tile of h (C-layout -> LDS row-major transpose)
#pragma unroll
    for (int j = 0; j < 8; ++j) {
      float v = h[j] > 0.0f ? h[j] : 0.0f;
      hst[(half * 8 + j) * HS + lm] = v;
    }
    // consume as A-fragments of GEMM2b K-chunks 4*n2 .. 4*n2+3
#pragma unroll
    for (int c2 = 0; c2 < 4; ++c2) {
      v2f ha = *(const v2f*)&hst[lm * HS + 4 * c2 + 2 * half];
      int kg = n2 * 16 + 4 * c2 + 2 * half;
#pragma unroll
      for (int n = 0; n < 4; ++n) {
        int nc = n * 16 + lm;
        v2f b2 = { lds[OFF_W2BF + kg * W2BS + nc],
                   lds[OFF_W2BF + (kg + 1) * W2BS + nc] };
        acc2[n] = wmma4(ha, b2, acc2[n]);
      }
    }
  }

  // --- global max-pool over x2 (relu'ed, >= 0)
#pragma unroll
  for (int n = 0; n < 4; ++n) {
    float mx = 0.0f;
#pragma unroll
    for (int j = 0; j < 8; ++j) {
      float v = acc2[n][j] > 0.0f ? acc2[n][j] : 0.0f;
      if ((FULL || rv[j]) && v > mx) mx = v;
    }
    atomicMax((int*)gmax + n * 16 + lm, __float_as_int(mx));
  }
}

// ---------------- fused main kernel ----------------
__global__ __launch_bounds__(256) void pn_main(
    const float* __restrict__ in, const int* __restrict__ seg,
    const float* __restrict__ W1,  const float* __restrict__ g1,
    const float* __restrict__ b1,  const float* __restrict__ m1,
    const float* __restrict__ v1,
    const float* __restrict__ W2a, const float* __restrict__ g2a,
    const float* __restrict__ b2a, const float* __restrict__ m2a,
    const float* __restrict__ v2a,
    const float* __restrict__ W2b, const float* __restrict__ g2b,
    const float* __restrict__ b2b, const float* __restrict__ m2b,
    const float* __restrict__ v2b,
    float* __restrict__ out, float* __restrict__ gmax,
    int N, int ntiles)
{
  __shared__ float lds[LDS_TOTAL];
  const int tid = threadIdx.x;

  // --- fold BN scales/biases (scratch in h-staging area, consumed pre-barrier2)
  float* scr1  = &lds[OFF_HST];        // 64
  float* scr2a = scr1 + 64;            // 128
  float* scr2b = scr2a + 128;          // 64
  if (tid < 64) {
    float s = g1[tid] * rsqrtf(v1[tid] + PN_EPS);
    scr1[tid] = s;  lds[OFF_B1F + tid] = b1[tid] - m1[tid] * s;
  }
  if (tid < 128) {
    float s = g2a[tid] * rsqrtf(v2a[tid] + PN_EPS);
    scr2a[tid] = s; lds[OFF_B2AF + tid] = b2a[tid] - m2a[tid] * s;
  }
  if (tid >= 128 && tid < 192) {
    int c = tid - 128;
    float s = g2b[c] * rsqrtf(v2b[c] + PN_EPS);
    scr2b[c] = s;   lds[OFF_B2BF + c] = b2b[c] - m2b[c] * s;
  }
  __syncthreads();
  for (int i = tid; i < 16 * 64;  i += 256) { int k = i >> 6, c = i & 63;
    lds[OFF_W1F  + k * W1S  + c] = W1[i]  * scr1[c];  }
  for (int i = tid; i < 16 * 128; i += 256) { int k = i >> 7, c = i & 127;
    lds[OFF_W2AF + k * W2AS + c] = W2a[i] * scr2a[c]; }
  for (int i = tid; i < 128 * 64; i += 256) { int k = i >> 6, c = i & 63;
    lds[OFF_W2BF + k * W2BS + c] = W2b[i] * scr2b[c]; }
  __syncthreads();

  const int wave = tid >> 5, lane = tid & 31;
  const int half = lane >> 4, lm = lane & 15;
  const int t = blockIdx.x * 8 + wave;
  if (t >= ntiles) return;
  const int r0 = t * 16;
  float* hst = &lds[OFF_HST + wave * 16 * HS];

  if (r0 + 16 <= N) {   // wave-uniform: hot path, no per-row guards
    pn_tile<true >(in, seg, out, gmax, lds, hst, r0, N, lane, half, lm);
  } else {              // tail tile (only if N % 16 != 0)
    pn_tile<false>(in, seg, out, gmax, lds, hst, r0, N, lane, half, lm);
  }
}

// ---------------- finalize: broadcast max-pool into group 2 ----------------
__global__ void pn_fin(float* __restrict__ out, const float* __restrict__ gmax, int S) {
  int i = blockIdx.x * blockDim.x + threadIdx.x;
  if (i < S * 64) {
    int c = i & 63, s = i >> 6;
    __builtin_nontemporal_store(gmax[c], out + s * 192 + c * 3 + 2);
  }
}

extern "C" void kernel_launch(void* const* d_in, const int* in_sizes, int n_in,
                              void* d_out, int out_size, void* d_ws, size_t ws_size,
                              hipStream_t stream) {
  const float* in  = (const float*)d_in[0];
  const int*   seg = (const int*)  d_in[1];
  // d_in[2] = num_segments (derived from out_size instead; no device read needed)
  const float* W1  = (const float*)d_in[3];
  const float* g1  = (const float*)d_in[4];
  const float* b1  = (const float*)d_in[5];
  const float* m1  = (const float*)d_in[6];
  const float* v1  = (const float*)d_in[7];
  const float* W2a = (const float*)d_in[8];
  const float* g2a = (const float*)d_in[9];
  const float* b2a = (const float*)d_in[10];
  const float* m2a = (const float*)d_in[11];
  const float* v2a = (const float*)d_in[12];
  const float* W2b = (const float*)d_in[13];
  const float* g2b = (const float*)d_in[14];
  const float* b2b = (const float*)d_in[15];
  const float* m2b = (const float*)d_in[16];
  const float* v2b = (const float*)d_in[17];

  float* out  = (float*)d_out;
  float* gmax = (float*)d_ws;           // 64 floats of scratch

  const int N      = in_sizes[1];       // number of points
  const int S      = out_size / 192;    // number of segments
  const int ntiles = (N + 15) / 16;

  pn_zero<<<(out_size + 255) / 256, 256, 0, stream>>>(out, out_size, gmax);
  pn_main<<<(ntiles + 7) / 8, 256, 0, stream>>>(
      in, seg, W1, g1, b1, m1, v1, W2a, g2a, b2a, m2a, v2a,
      W2b, g2b, b2b, m2b, v2b, out, gmax, N, ntiles);
  pn_fin<<<(S * 64 + 255) / 256, 256, 0, stream>>>(out, gmax, S);
}